// GNN_68298569941641
// MI455X (gfx1250) — compile-verified
//
#include <hip/hip_runtime.h>

typedef float v2f __attribute__((ext_vector_type(2)));
typedef float v8f __attribute__((ext_vector_type(8)));

#define NEGV -1.0e9f
#define HSTR 84   // h / g LDS row stride (avoids 2-way bank conflicts on D stores)

// shared (block) LDS layout, in floats
#define SH_WPAD 0          // 8 x 80, W padded (K pad 7->8, N pad 72->80)
#define SH_AS   640        // 72
#define SH_AD   712        // 72
#define SH_MASK 784        // 36*36 adjacency mask as float
#define SH_WDWN 2080       // 72
#define SH_WMU  2152       // 72*10
#define SH_BMU  2872       // 10
#define SH_BDWN 2882       // 1
#define SH_TOTAL 2883
// per-wave LDS layout, in floats
#define PW_XCP  0          // 36 x 9 (k=0..6 data, k=7 zero pad)
#define PW_T    324        // 36
#define PW_HS   360        // 36 x HSTR (cols 0..71 h, 72..79 zero, 80..83 unused)
#define PW_ES   3384       // 8 x 36
#define PW_ED   3672       // 8 x 36
#define PW_ATT  3960       // 36 x 36
#define PW_GS   5256       // 36 x HSTR (elu(out))
#define PW_DOWN 8280       // 36
#define PW_TOTAL 8316

__global__ __launch_bounds__(128) void gat_mu_kernel(
    const float* __restrict__ x, const int* __restrict__ adj,
    const float* __restrict__ W, const float* __restrict__ a_src,
    const float* __restrict__ a_dst, const float* __restrict__ W_down,
    const float* __restrict__ b_down, const float* __restrict__ W_mu,
    const float* __restrict__ b_mu, float* __restrict__ out, int Bn)
{
    extern __shared__ float smem[];
    const int tid  = threadIdx.x;
    const int wave = tid >> 5;
    const int lane = tid & 31;
    const int lh   = lane >> 4;    // half-wave select (K split for A/B operands)
    const int ll   = lane & 15;    // M (A) / N (B,C,D) index within tile

    float* Wpad  = smem + SH_WPAD;
    float* aS    = smem + SH_AS;
    float* aD    = smem + SH_AD;
    float* maskS = smem + SH_MASK;
    float* Wdwn  = smem + SH_WDWN;
    float* Wmu   = smem + SH_WMU;
    float* bmuS  = smem + SH_BMU;
    float* bdwnS = smem + SH_BDWN;
    float* pw    = smem + SH_TOTAL + wave * PW_TOTAL;
    float* xcp   = pw + PW_XCP;
    float* tS    = pw + PW_T;
    float* hS    = pw + PW_HS;
    float* eS    = pw + PW_ES;
    float* eD    = pw + PW_ED;
    float* attnS = pw + PW_ATT;
    float* gS    = pw + PW_GS;
    float* downS = pw + PW_DOWN;

    // ---- stage 0: block-shared weights / adjacency into LDS ----
    for (int i = tid; i < 640; i += 128) {
        int k = i / 80, c = i % 80;
        Wpad[i] = (k < 7 && c < 72) ? W[(c / 9) * 63 + k * 9 + (c % 9)] : 0.f;
    }
    for (int i = tid; i < 72; i += 128) {
        aS[i] = a_src[i]; aD[i] = a_dst[i]; Wdwn[i] = W_down[i];
    }
    for (int i = tid; i < 1296; i += 128) maskS[i] = (adj[i] > 0) ? 1.f : 0.f;
    for (int i = tid; i < 720;  i += 128) Wmu[i] = W_mu[i];
    if (tid < 10) bmuS[tid] = b_mu[tid];
    if (tid == 0) bdwnS[0] = b_down[0];
    __syncthreads();

    const int  b      = blockIdx.x * 4 + wave;
    const bool active = (b < Bn);

    // ---- stage 1a: load x[b], build xc (drop col 3) and t = x[:,4] ----
    if (active) {
        const float* xb = x + (size_t)b * 288;
        for (int idx = lane; idx < 288; idx += 32) {
            int n = idx >> 3, f = idx & 7;
            float v = xb[idx];
            if (f < 3) xcp[n * 9 + f] = v;
            else if (f >= 4) { xcp[n * 9 + f - 1] = v; if (f == 4) tS[n] = v; }
        }
        for (int n = lane; n < 36; n += 32) xcp[n * 9 + 7] = 0.f;  // K pad
    }
    __syncthreads();

    // ---- stage 1b: h = xc(36x8) @ Wpad(8x80) via WMMA f32 16x16x4 ----
    if (active) {
        for (int mt = 0; mt < 3; ++mt) {
            int m = ll + 16 * mt;
            int msafe = (m < 36) ? m : 0;           // clamp: garbage only hits ignored D rows
            for (int nt = 0; nt < 5; ++nt) {
                v8f acc = {};
#pragma unroll
                for (int ks = 0; ks < 2; ++ks) {
                    int kb = ks * 4 + 2 * lh;
                    v2f av, bv;
                    av.x = xcp[msafe * 9 + kb];
                    av.y = xcp[msafe * 9 + kb + 1];
                    bv.x = Wpad[kb * 80 + nt * 16 + ll];
                    bv.y = Wpad[(kb + 1) * 80 + nt * 16 + ll];
                    acc = __builtin_amdgcn_wmma_f32_16x16x4_f32(
                        false, av, false, bv, (short)0, acc, false, false);
                }
#pragma unroll
                for (int r = 0; r < 8; ++r) {
                    int m2 = r + 8 * lh + 16 * mt;
                    if (m2 < 36) hS[m2 * HSTR + nt * 16 + ll] = acc[r];
                }
            }
        }
    }
    __syncthreads();

    // ---- stage 1c: e_src / e_dst ----
    if (active) {
        for (int idx = lane; idx < 576; idx += 32) {
            int which = (idx >= 288);
            int rem = which ? idx - 288 : idx;
            int hh = rem / 36, n = rem - hh * 36;
            const float* av = which ? aD : aS;
            float s = 0.f;
#pragma unroll
            for (int d = 0; d < 9; ++d) s += hS[n * HSTR + hh * 9 + d] * av[hh * 9 + d];
            if (which) eD[hh * 36 + n] = s; else eS[hh * 36 + n] = s;
        }
    }
    __syncthreads();

    // ---- per-head: masked leaky-relu softmax, then attn @ h via WMMA ----
    for (int hh = 0; hh < 8; ++hh) {
        if (active) {
            for (int i = lane; i < 36; i += 32) {
                float es = eS[hh * 36 + i];
                float mx = -3.0e38f;
                for (int j = 0; j < 36; ++j) {
                    float e = es + eD[hh * 36 + j];
                    e = (e > 0.f) ? e : 0.2f * e;
                    e = (maskS[i * 36 + j] > 0.f) ? e : NEGV;
                    mx = fmaxf(mx, e);
                }
                float sum = 0.f;
                for (int j = 0; j < 36; ++j) {
                    float e = es + eD[hh * 36 + j];
                    e = (e > 0.f) ? e : 0.2f * e;
                    e = (maskS[i * 36 + j] > 0.f) ? e : NEGV;
                    float ex = __expf(e - mx);
                    sum += ex;
                    attnS[i * 36 + j] = ex;
                }
                float inv = 1.f / sum;
                for (int j = 0; j < 36; ++j) attnS[i * 36 + j] *= inv;
            }
        }
        __syncthreads();

        if (active) {
            for (int mt = 0; mt < 3; ++mt) {
                int m = ll + 16 * mt;
                int msafe = (m < 36) ? m : 0;
                v8f acc = {};
#pragma unroll
                for (int ks = 0; ks < 9; ++ks) {     // K = 36 = 9 x 4
                    int kb = ks * 4 + 2 * lh;
                    v2f av, bv;
                    av.x = attnS[msafe * 36 + kb];
                    av.y = attnS[msafe * 36 + kb + 1];
                    bv.x = hS[kb * HSTR + hh * 9 + ll];
                    bv.y = hS[(kb + 1) * HSTR + hh * 9 + ll];
                    acc = __builtin_amdgcn_wmma_f32_16x16x4_f32(
                        false, av, false, bv, (short)0, acc, false, false);
                }
#pragma unroll
                for (int r = 0; r < 8; ++r) {
                    int m2 = r + 8 * lh + 16 * mt;
                    if (m2 < 36 && ll < 9) {
                        float v = acc[r];
                        v = (v > 0.f) ? v : (__expf(v) - 1.f);   // elu
                        gS[m2 * HSTR + hh * 9 + ll] = v;
                    }
                }
            }
        }
        __syncthreads();
    }

    // ---- down-projection (72 -> 1) ----
    if (active) {
        for (int n = lane; n < 36; n += 32) {
            float s = bdwnS[0];
#pragma unroll 8
            for (int c = 0; c < 72; ++c) s += gS[n * HSTR + c] * Wdwn[c];
            downS[n] = s;
        }
    }
    __syncthreads();

    // ---- feat = leaky(concat(down, t)); mu = feat @ W_mu + b_mu ----
    if (active && lane < 10) {
        int a = lane;
        float acc = bmuS[a];
        for (int n = 0; n < 36; ++n) {
            float dv = downS[n]; dv = (dv > 0.f) ? dv : 0.01f * dv;
            float tv = tS[n];    tv = (tv > 0.f) ? tv : 0.01f * tv;
            acc += dv * Wmu[n * 10 + a] + tv * Wmu[(36 + n) * 10 + a];
        }
        out[(size_t)b * 10 + a] = acc;
    }
}

extern "C" void kernel_launch(void* const* d_in, const int* in_sizes, int n_in,
                              void* d_out, int out_size, void* d_ws, size_t ws_size,
                              hipStream_t stream) {
    const float* x      = (const float*)d_in[0];
    const int*   adj    = (const int*)  d_in[1];
    const float* W      = (const float*)d_in[2];
    const float* a_src  = (const float*)d_in[3];
    const float* a_dst  = (const float*)d_in[4];
    const float* W_down = (const float*)d_in[5];
    const float* b_down = (const float*)d_in[6];
    const float* W_mu   = (const float*)d_in[7];
    const float* b_mu   = (const float*)d_in[8];
    float* out = (float*)d_out;

    int Bn = in_sizes[0] / 288;                 // 8192
    int blocks = (Bn + 3) / 4;                  // 4 waves (batches) per 128-thread block
    size_t shbytes = (size_t)(SH_TOTAL + 4 * PW_TOTAL) * sizeof(float);
    gat_mu_kernel<<<dim3(blocks), dim3(128), shbytes, stream>>>(
        x, adj, W, a_src, a_dst, W_down, b_down, W_mu, b_mu, out, Bn);
}